// Net_7825430413940
// MI455X (gfx1250) — compile-verified
//
#include <hip/hip_runtime.h>
#include <hip/hip_bf16.h>

// Problem constants (from the reference)
#define G_ 64
#define N_ 2048
#define E_ 32768
#define D_ 128
#define H_ 128
#define O_ 2
#define L_ 3

// LDS padded strides (floats)
#define A_STRIDE 132   // 16 rows, +4 pad -> rows hit distinct banks for column reads
#define W_STRIDE 130   // TDM pad: 2 DWORDs after every 128 -> kills lo/hi-half conflicts

typedef __attribute__((ext_vector_type(2))) float v2f;
typedef __attribute__((ext_vector_type(4))) float v4f;
typedef __attribute__((ext_vector_type(8))) float v8f;
typedef __attribute__((ext_vector_type(4))) unsigned int v4u;
typedef __attribute__((ext_vector_type(4))) int v4i;
typedef __attribute__((ext_vector_type(8))) int v8i;

// ---------------------------------------------------------------------------
// Utility: zero-fill (float4 granularity)
// ---------------------------------------------------------------------------
__global__ void zero_kernel(float* __restrict__ p, long long n4) {
    long long i = (long long)blockIdx.x * blockDim.x + threadIdx.x;
    if (i < n4) {
        v4f z = {};
        ((v4f*)p)[i] = z;
    }
}

// ---------------------------------------------------------------------------
// Gather: xcur[g,n,:] = all_features[feature_index[g,n],:]   (float4 per lane)
// ---------------------------------------------------------------------------
__global__ void gather_kernel(const float* __restrict__ allf,
                              const int* __restrict__ fidx,
                              float* __restrict__ xcur) {
    long long t = (long long)blockIdx.x * blockDim.x + threadIdx.x;
    long long row = t >> 5;          // 32 float4 chunks per 128-float row
    int c4 = (int)(t & 31);
    if (row >= (long long)G_ * N_) return;
    int fi = fidx[row];
    const v4f* src = (const v4f*)(allf + (long long)fi * D_);
    v4f* dst = (v4f*)(xcur + row * (long long)D_);
    dst[c4] = src[c4];
}

// ---------------------------------------------------------------------------
// Degree / normalization:  dinv starts at 1 (self-loop), += indegree, -> rsqrt
// ---------------------------------------------------------------------------
__global__ void dinv_init_kernel(float* __restrict__ dinv) {
    int i = blockIdx.x * blockDim.x + threadIdx.x;
    if (i < G_ * N_) dinv[i] = 1.0f;
}

__global__ void deg_kernel(const int* __restrict__ eidx, float* __restrict__ dinv) {
    int g = blockIdx.y;
    int e = blockIdx.x * blockDim.x + threadIdx.x;
    if (e >= E_) return;
    int d = eidx[(long long)g * 2 * E_ + E_ + e];   // dst row
    atomicAdd(&dinv[g * N_ + d], 1.0f);
}

__global__ void rsqrt_kernel(float* __restrict__ dinv) {
    int i = blockIdx.x * blockDim.x + threadIdx.x;
    if (i < G_ * N_) dinv[i] = rsqrtf(dinv[i]);
}

// ---------------------------------------------------------------------------
// WMMA fp32 matmul:  Y[g] = op( (X[g] (+Z[g])) @ W + bias )
//   X,Z,Y : [G, N_, H_]   W : [H_, H_] row-major (K x N)   bias : [H_]
//   Block = 256 threads = 8 wave32s -> one 16-row strip x all 8 col-tiles.
//   A strip (16x128, +Z fused) staged in LDS by all threads.
//   W (128x128) staged in LDS by the Tensor Data Mover with row padding.
//   Inner loop: ds_load A pair + ds_load B pair + V_WMMA_F32_16X16X4_F32.
//   Optional fused mean-pool accumulation into pooled[g, H_].
// ---------------------------------------------------------------------------
template <bool HAS_Z>
__global__ void __launch_bounds__(256)
mm_wmma_kernel(const float* __restrict__ X,
               const float* __restrict__ Z,
               const float* __restrict__ W,
               const float* __restrict__ bias,
               float* __restrict__ Y,
               float* __restrict__ pooled,
               const int* __restrict__ action,
               int layer, int relu) {
    __shared__ float As[16 * A_STRIDE];    //  8448 B
    __shared__ float Ws[H_ * W_STRIDE];    // 66560 B (TDM writes padded rows)

    if (layer >= 0 && layer > *action) return;   // uniform skip (inactive layer)

    const int g     = blockIdx.y;
    const int tileM = blockIdx.x;                // 0..127 -> rows of 16
    const int tid   = threadIdx.x;
    const int lane  = tid & 31;
    const int wv    = tid >> 5;                  // tileN = wv (0..7)

    const long long base = (long long)g * N_ * H_;
    const float* Xg = X + base + (long long)tileM * 16 * H_;
    const float* Zg = HAS_Z ? (Z + base + (long long)tileM * 16 * H_) : nullptr;

    // ---- stage A strip (16 x 128) into LDS, fusing the +Z residual add ----
    for (int i = tid; i < 16 * 32; i += 256) {   // 512 float4 chunks
        int r = i >> 5, c4 = (i & 31) * 4;
        v4f a = *(const v4f*)(Xg + (long long)r * H_ + c4);
        if (HAS_Z) {
            v4f z = *(const v4f*)(Zg + (long long)r * H_ + c4);
            a.x += z.x; a.y += z.y; a.z += z.z; a.w += z.w;
        }
        *(v4f*)(&As[r * A_STRIDE + c4]) = a;
    }

    // ---- stage W (128x128) into LDS via the Tensor Data Mover (wave 0) ----
    if (wv == 0) {
        unsigned long long ga = (unsigned long long)(uintptr_t)W;
        unsigned lds = (unsigned)(uintptr_t)(void*)Ws;   // low 32 bits = LDS offset
        // D# group 0: count=1 | lds_addr | global_addr | type=2
        v4u g0 = { 1u,
                   lds,
                   (unsigned)(ga & 0xFFFFFFFFu),
                   (unsigned)((ga >> 32) & 0x01FFFFFFu) | 0x80000000u };
        // D# group 1: data_size=4B, pad_enable, pad_interval=128DW, pad_amount=2DW,
        //             tensor_dim0/1 = 128, tile_dim0/1 = 128, dim0_stride = 128
        v8i g1 = { (int)((2u << 16) | (1u << 20) | (6u << 22) | (1u << 25)),
                   (int)(128u << 16),   // [63:48]  tensor_dim0 lo16
                   (int)(128u << 16),   // [95:80]  tensor_dim1 lo16
                   (int)(128u << 16),   // [127:112] tile_dim0
                   (int)128,            // [143:128] tile_dim1 (tile_dim2 = 0)
                   (int)128,            // [191:160] tensor_dim0_stride lo32
                   0, 0 };
        v4i zz4 = {};
        v8i zz8 = {};
        __builtin_amdgcn_tensor_load_to_lds(g0, g1, zz4, zz4, zz8, 0);
        __builtin_amdgcn_s_wait_tensorcnt(0);
    }
    __syncthreads();

    // ---- 16x16 output tile per wave, K swept in steps of 4 ----
    const int  l15  = lane & 15;
    const bool hi   = lane >= 16;
    const int  koff = hi ? 2 : 0;
    const int  colB = wv * 16 + l15;

    v8f acc = {};
#pragma unroll 8
    for (int k = 0; k < H_; k += 4) {
        const int ka = k + koff;
        // A (16x4): lanes 0-15 -> K=k,k+1 ; lanes 16-31 -> K=k+2,k+3
        v2f a = *(const v2f*)(&As[l15 * A_STRIDE + ka]);
        // B (4x16): VGPR0 = rows k / k+2, VGPR1 = rows k+1 / k+3
        v2f b;
        b.x = Ws[ka * W_STRIDE + colB];
        b.y = Ws[(ka + 1) * W_STRIDE + colB];
        acc = __builtin_amdgcn_wmma_f32_16x16x4_f32(
            /*neg_a=*/false, a, /*neg_b=*/false, b,
            /*c_mod=*/(short)0, acc, /*reuse_a=*/false, /*reuse_b=*/false);
    }

    const float bval  = bias ? bias[colB] : 0.0f;
    const int   mbase = tileM * 16 + (hi ? 8 : 0);
    float* Yg = Y ? Y + base : nullptr;
    float psum = 0.0f;
#pragma unroll
    for (int j = 0; j < 8; ++j) {
        float v = acc[j] + bval;
        if (relu) v = fmaxf(v, 0.0f);
        if (Yg) Yg[(long long)(mbase + j) * H_ + colB] = v;
        psum += v;
    }
    if (pooled) {
        // lanes L and L+16 hold the same column (different row halves)
        psum += __shfl_xor(psum, 16, 32);
        if (!hi) atomicAdd(&pooled[g * H_ + colB], psum);
    }
}

// ---------------------------------------------------------------------------
// Edge scatter: xnext[g,dst,:] += h[g,src,:] * dinv[src]*dinv[dst]
//   One warp per edge; each lane handles 4 contiguous columns (float4 load).
// ---------------------------------------------------------------------------
__global__ void __launch_bounds__(256)
scatter_kernel(const int* __restrict__ eidx,
               const float* __restrict__ h,
               const float* __restrict__ dinv,
               float* __restrict__ xnext,
               const int* __restrict__ action, int layer) {
    if (layer > *action) return;
    const int g    = blockIdx.y;
    const int e    = (int)(((long long)blockIdx.x * blockDim.x + threadIdx.x) >> 5);
    const int lane = threadIdx.x & 31;
    if (e >= E_) return;

    const int* eg = eidx + (long long)g * 2 * E_;
    const int s = eg[e];
    const int d = eg[E_ + e];
    const float c = dinv[g * N_ + s] * dinv[g * N_ + d];

    const float* hs = h     + ((long long)g * N_ + s) * H_ + lane * 4;
    float*       xd = xnext + ((long long)g * N_ + d) * H_ + lane * 4;
    v4f v = *(const v4f*)hs;
    atomicAdd(xd + 0, v.x * c);
    atomicAdd(xd + 1, v.y * c);
    atomicAdd(xd + 2, v.z * c);
    atomicAdd(xd + 3, v.w * c);
}

// ---------------------------------------------------------------------------
// Finalize layer: xcur = relu( xnext + h * dinv^2 (self-loop) + bias )
// ---------------------------------------------------------------------------
__global__ void finalize_kernel(const float* __restrict__ xnext,
                                const float* __restrict__ h,
                                const float* __restrict__ dinv,
                                const float* __restrict__ bias,
                                float* __restrict__ xcur,
                                const int* __restrict__ action, int layer) {
    if (layer > *action) return;
    long long t = (long long)blockIdx.x * blockDim.x + threadIdx.x;
    if (t >= (long long)G_ * N_ * H_) return;
    long long row = t / H_;
    int col = (int)(t - row * H_);
    float di = dinv[row];
    float v = xnext[t] + h[t] * di * di + bias[col];
    xcur[t] = fmaxf(v, 0.0f);
}

// ---------------------------------------------------------------------------
// Head: logits = mean_feats @ lin_w + lin_b ; out = [log_softmax | logits]
// ---------------------------------------------------------------------------
__global__ void head_kernel(const float* __restrict__ pooled,
                            const float* __restrict__ lin_w,
                            const float* __restrict__ lin_b,
                            float* __restrict__ out) {
    int g = threadIdx.x;
    if (g >= G_) return;
    float m0 = lin_b[0], m1 = lin_b[1];
    const float inv = 1.0f / (float)N_;
    for (int j = 0; j < H_; ++j) {
        float p = pooled[g * H_ + j] * inv;
        m0 += p * lin_w[j * O_ + 0];
        m1 += p * lin_w[j * O_ + 1];
    }
    float mx  = fmaxf(m0, m1);
    float lse = mx + logf(__expf(m0 - mx) + __expf(m1 - mx));
    out[g * O_ + 0] = m0 - lse;            // log_softmax
    out[g * O_ + 1] = m1 - lse;
    out[G_ * O_ + g * O_ + 0] = m0;        // raw logits
    out[G_ * O_ + g * O_ + 1] = m1;
}

// ---------------------------------------------------------------------------
// Launcher
// ---------------------------------------------------------------------------
extern "C" void kernel_launch(void* const* d_in, const int* in_sizes, int n_in,
                              void* d_out, int out_size, void* d_ws, size_t ws_size,
                              hipStream_t stream) {
    (void)in_sizes; (void)n_in; (void)out_size; (void)ws_size;

    const float* allf      = (const float*)d_in[0];
    const int*   fidx      = (const int*)d_in[1];
    const int*   eidx      = (const int*)d_in[2];
    const int*   action    = (const int*)d_in[3];
    const float* lin_res_w = (const float*)d_in[4];
    const float* lin_res_b = (const float*)d_in[5];
    const float* gcn_w     = (const float*)d_in[6];
    const float* gcn_b     = (const float*)d_in[7];
    const float* fc1_w     = (const float*)d_in[8];
    const float* fc1_b     = (const float*)d_in[9];
    const float* lin_w     = (const float*)d_in[10];
    const float* lin_b     = (const float*)d_in[11];

    // Workspace layout (floats): xcur | x1 | h | xnext | dinv | pooled
    const size_t npg = (size_t)G_ * N_ * H_;          // 16.78M floats each
    float* xcur   = (float*)d_ws;
    float* x1     = xcur   + npg;
    float* h      = x1     + npg;
    float* xnext  = h      + npg;
    float* dinv   = xnext  + npg;
    float* pooled = dinv   + (size_t)G_ * N_;

    // 1) gather node features
    {
        long long threads = (long long)G_ * N_ * 32;
        gather_kernel<<<(unsigned)((threads + 255) / 256), 256, 0, stream>>>(allf, fidx, xcur);
    }

    // 2) residual branch: x1 = relu(feats @ lin_res_w + b)
    dim3 mmGrid(N_ / 16, G_);
    mm_wmma_kernel<false><<<mmGrid, 256, 0, stream>>>(xcur, nullptr, lin_res_w, lin_res_b,
                                                      x1, nullptr, action, -1, 1);

    // 3) symmetric-normalization coefficients
    dinv_init_kernel<<<(G_ * N_ + 255) / 256, 256, 0, stream>>>(dinv);
    deg_kernel<<<dim3((E_ + 255) / 256, G_), 256, 0, stream>>>(eidx, dinv);
    rsqrt_kernel<<<(G_ * N_ + 255) / 256, 256, 0, stream>>>(dinv);

    // 4) GCN layers (kernels self-skip when layer > *action)
    for (int k = 0; k < L_; ++k) {
        mm_wmma_kernel<false><<<mmGrid, 256, 0, stream>>>(xcur, nullptr,
                                                          gcn_w + (size_t)k * H_ * H_,
                                                          nullptr, h, nullptr, action, k, 0);
        long long n4 = (long long)(npg / 4);
        zero_kernel<<<(unsigned)((n4 + 255) / 256), 256, 0, stream>>>(xnext, n4);
        scatter_kernel<<<dim3((E_ * 32) / 256, G_), 256, 0, stream>>>(eidx, h, dinv,
                                                                      xnext, action, k);
        finalize_kernel<<<(unsigned)((npg + 255) / 256), 256, 0, stream>>>(
            xnext, h, dinv, gcn_b + (size_t)k * H_, xcur, action, k);
    }

    // 5) final: pooled[g] = sum_n relu((xcur + x1) @ fc1_w + b)   (fused pool)
    zero_kernel<<<(G_ * H_ / 4 + 255) / 256, 256, 0, stream>>>(pooled, G_ * H_ / 4);
    mm_wmma_kernel<true><<<mmGrid, 256, 0, stream>>>(xcur, x1, fc1_w, fc1_b,
                                                     nullptr, pooled, action, -1, 1);

    // 6) logits + log_softmax
    head_kernel<<<1, G_, 0, stream>>>(pooled, lin_w, lin_b, (float*)d_out);
}